// DualLoss_32581621908064
// MI455X (gfx1250) — compile-verified
//
#include <hip/hip_runtime.h>
#include <math.h>

typedef __attribute__((ext_vector_type(2))) float v2f;
typedef __attribute__((ext_vector_type(8))) float v8f;

#define B_   8
#define M_   16
#define S_   100
#define N_   2048
#define SP   112        // S padded to 7 tiles of 16
#define RT_  7          // row tiles over S
#define CT_  128        // col tiles over N
#define EPSF 1e-6f
#define BIGF 1e30f

__device__ __forceinline__ float fexp_f(float x, float p) {
    float s = (x > 0.f) ? 1.f : ((x < 0.f) ? -1.f : 0.f);
    return s * powf(fabsf(x) + EPSF, p);
}

// ---------------------------------------------------------------------------
// Kernel 1: one workgroup per (b,m). Stages surface points P[112][4] and
// transformed cloud Q[2048][4] in LDS, then computes the 100x2048 squared
// distance matrix with V_WMMA_F32_16X16X4_F32 (dist = |p|^2+|q|^2-2 p.q).
// Row-min over N is reduced with fire-and-forget ds_min_u32 atomics on the
// ordered-uint encoding (valid: values clamped >= 0); column-min over S needs
// only one lane-16 swap. Emits dmin_ws[b][n][m] and meanS_ws[b][m].
// ---------------------------------------------------------------------------
__global__ __launch_bounds__(256) void sq_dist_kernel(
    const float* __restrict__ pcl,        // [B][N][3]
    const float* __restrict__ trans,      // [B][M][3]
    const float* __restrict__ rot,        // [B][M][4]
    const float* __restrict__ size,       // [B][M][3]
    const float* __restrict__ shape_eps,  // [B][M][2]
    const float* __restrict__ deform,     // [B][M][2]
    const float* __restrict__ etas,       // [S]
    const float* __restrict__ omegas,     // [S]
    float* __restrict__ dmin_ws,          // [B][N][M]
    float* __restrict__ meanS_ws)         // [B][M]
{
    __shared__ float P[SP * 4];           // surface points, K-padded
    __shared__ float pp[SP];              // |p|^2  (BIGF for padded rows)
    __shared__ float Q[N_ * 4];           // transformed cloud, K-padded
    __shared__ float qq[N_];              // |q|^2
    __shared__ unsigned int rowminU[SP];  // min over n, ordered-uint encoding

    const int wg   = blockIdx.x;          // 0..B*M-1
    const int b    = wg / M_;
    const int m    = wg % M_;
    const int tid  = threadIdx.x;
    const int lane = tid & 31;
    const int wave = tid >> 5;

    // Pull the point-cloud stream toward the caches early (global_prefetch_b8).
    __builtin_prefetch(pcl + (size_t)b * N_ * 3 + tid * 16, 0, 0);

    // --- rotation matrix from quaternion (redundantly per thread) ---
    const float* qp = rot + (size_t)(b * M_ + m) * 4;
    float qw = qp[0], qx = qp[1], qy = qp[2], qz = qp[3];
    float qn = sqrtf(qw * qw + qx * qx + qy * qy + qz * qz) + EPSF;
    qw /= qn; qx /= qn; qy /= qn; qz /= qn;
    const float R00 = 1.f - 2.f * (qy * qy + qz * qz);
    const float R01 = 2.f * (qx * qy - qw * qz);
    const float R02 = 2.f * (qx * qz + qw * qy);
    const float R10 = 2.f * (qx * qy + qw * qz);
    const float R11 = 1.f - 2.f * (qx * qx + qz * qz);
    const float R12 = 2.f * (qy * qz - qw * qx);
    const float R20 = 2.f * (qx * qz - qw * qy);
    const float R21 = 2.f * (qy * qz + qw * qx);
    const float R22 = 1.f - 2.f * (qx * qx + qy * qy);
    const float tx = trans[(b * M_ + m) * 3 + 0];
    const float ty = trans[(b * M_ + m) * 3 + 1];
    const float tz = trans[(b * M_ + m) * 3 + 2];

    // --- superquadric surface points into LDS ---
    if (tid < SP) {
        float px = 0.f, py = 0.f, pz = 0.f, n2 = BIGF;  // pad rows: huge |p|^2
        if (tid < S_) {
            const float e1 = shape_eps[(b * M_ + m) * 2 + 0];
            const float e2 = shape_eps[(b * M_ + m) * 2 + 1];
            const float a1 = size[(b * M_ + m) * 3 + 0];
            const float a2 = size[(b * M_ + m) * 3 + 1];
            const float a3 = size[(b * M_ + m) * 3 + 2];
            const float d1 = deform[(b * M_ + m) * 2 + 0];
            const float d2 = deform[(b * M_ + m) * 2 + 1];
            float eta = etas[tid], om = omegas[tid];
            float ce = fexp_f(cosf(eta), e1);
            float se = fexp_f(sinf(eta), e1);
            float cw = fexp_f(cosf(om),  e2);
            float sw = fexp_f(sinf(om),  e2);
            float x = a1 * ce * cw;
            float y = a2 * ce * sw;
            float z = a3 * se;
            float fx = d1 * z / a3 + 1.f;
            float fy = d2 * z / a3 + 1.f;
            px = fx * x; py = fy * y; pz = z;
            n2 = px * px + py * py + pz * pz;
        }
        P[tid * 4 + 0] = px;
        P[tid * 4 + 1] = py;
        P[tid * 4 + 2] = pz;
        P[tid * 4 + 3] = 0.f;
        pp[tid] = n2;
        rowminU[tid] = 0x7f800000u;  // +inf
    }

    // --- transform cloud into LDS: q = R (pcl - t) ---
    for (int n = tid; n < N_; n += 256) {
        const float* pc = pcl + (size_t)(b * N_ + n) * 3;
        float dx = pc[0] - tx, dy = pc[1] - ty, dz = pc[2] - tz;
        float vx = R00 * dx + R01 * dy + R02 * dz;
        float vy = R10 * dx + R11 * dy + R12 * dz;
        float vz = R20 * dx + R21 * dy + R22 * dz;
        Q[n * 4 + 0] = vx;
        Q[n * 4 + 1] = vy;
        Q[n * 4 + 2] = vz;
        Q[n * 4 + 3] = 0.f;
        qq[n] = vx * vx + vy * vy + vz * vz;
    }
    __syncthreads();

    // --- prebuild all 7 A fragments (16x4 f32 layout, ISA 7.12.2) ---
    const int arow  = lane & 15;          // matrix row (and D column id)
    const int kb    = (lane >> 4) * 2;    // K pair this lane holds
    const int rbase = (lane >> 4) * 8;    // D rows this lane holds
    v2f A[RT_];
#pragma unroll
    for (int rt = 0; rt < RT_; ++rt) {
        int s = rt * 16 + arow;
        A[rt].x = P[s * 4 + kb];
        A[rt].y = P[s * 4 + kb + 1];
    }

    // --- each wave owns column tiles ct = wave, wave+8, ... ---
    for (int ct = wave; ct < CT_; ct += 8) {
        const int n = ct * 16 + arow;
        v2f Bf;
        Bf.x = Q[n * 4 + kb];
        Bf.y = Q[n * 4 + kb + 1];
        const float qv = qq[n];
        float colmin = BIGF;

#pragma unroll
        for (int rt = 0; rt < RT_; ++rt) {
            v8f C = {};
            v8f D = __builtin_amdgcn_wmma_f32_16x16x4_f32(
                /*neg_a=*/false, A[rt], /*neg_b=*/false, Bf,
                /*c_mod=*/(short)0, C, /*reuse_a=*/false, /*reuse_b=*/false);

#pragma unroll
            for (int i = 0; i < 8; ++i) {
                int s = rt * 16 + rbase + i;
                float val = pp[s] + qv - 2.f * D[i];
                val = fmaxf(val, 0.f);              // true sq-dist >= 0
                colmin = fminf(colmin, val);
                // fire-and-forget row-min: no return value, no wait needed
                atomicMin(&rowminU[s], __float_as_uint(val));
            }
        }
        // combine the two row-halves for the column min, store dmin
        colmin = fminf(colmin, __shfl_xor(colmin, 16, 32));
        if (lane < 16)
            dmin_ws[((size_t)b * N_ + n) * M_ + m] = colmin;
    }
    __syncthreads();

    if (tid == 0) {
        float sum = 0.f;
        for (int s = 0; s < S_; ++s) sum += __uint_as_float(rowminU[s]);
        meanS_ws[b * M_ + m] = sum / (float)S_;
    }
}

// ---------------------------------------------------------------------------
// Kernel 2: per (b,n) sort of the 16 primitive distances, cumprod weighting.
// ---------------------------------------------------------------------------
__global__ __launch_bounds__(256) void pcl2prim_kernel(
    const float* __restrict__ dmin_ws,  // [B][N][M]
    const float* __restrict__ probs,    // [B][M]
    float* __restrict__ acc)            // acc[0] += sum(distance*minprob)
{
    const int gid = blockIdx.x * 256 + threadIdx.x;  // 0..B*N-1
    float loss = 0.f;
    if (gid < B_ * N_) {
        const int b = gid / N_;
        float dv[M_], pv[M_];
#pragma unroll
        for (int m = 0; m < M_; ++m) {
            dv[m] = dmin_ws[(size_t)gid * M_ + m];
            pv[m] = probs[b * M_ + m];
        }
        // insertion sort ascending by distance (stable)
        for (int i = 1; i < M_; ++i) {
            float dk = dv[i], pk = pv[i];
            int j = i - 1;
            while (j >= 0 && dv[j] > dk) {
                dv[j + 1] = dv[j]; pv[j + 1] = pv[j]; --j;
            }
            dv[j + 1] = dk; pv[j + 1] = pk;
        }
        float negcp = 1.f;
#pragma unroll
        for (int m = 0; m < M_; ++m) {
            loss  += dv[m] * pv[m] * negcp;
            negcp *= (1.f - pv[m]);
        }
    }
    __shared__ float red[256];
    red[threadIdx.x] = loss;
    __syncthreads();
    for (int s = 128; s > 0; s >>= 1) {
        if (threadIdx.x < s) red[threadIdx.x] += red[threadIdx.x + s];
        __syncthreads();
    }
    if (threadIdx.x == 0) atomicAdd(&acc[0], red[0]);
}

// ---------------------------------------------------------------------------
// Kernel 3: area-weighted prim-to-pcl term. One block of B*M=128 threads.
// ---------------------------------------------------------------------------
__global__ __launch_bounds__(128) void prim2pcl_kernel(
    const float* __restrict__ size,   // [B][M][3]
    const float* __restrict__ meanS,  // [B][M]
    float* __restrict__ acc)          // acc[1] += sum(meanS*area_norm)
{
    const int tid = threadIdx.x;           // b*16+m
    const int b   = tid / M_;
    float s0 = size[tid * 3 + 0], s1 = size[tid * 3 + 1], s2 = size[tid * 3 + 2];
    float t = powf(s0 * s1, 1.6f) / 3.f + powf(s0 * s2, 1.6f) / 3.f +
              powf(s1 * s2, 1.6f) / 3.f;
    float area = 4.f * 3.14159265358979323846f * powf(t, 0.625f);

    __shared__ float ar[128];
    ar[tid] = area;
    __syncthreads();
    float sumb = 0.f;
    for (int mm = 0; mm < M_; ++mm) sumb += ar[b * M_ + mm];
    float area_norm = (float)M_ * area / sumb;
    float contrib = meanS[tid] * area_norm;

    __shared__ float red[128];
    red[tid] = contrib;
    __syncthreads();
    for (int s = 64; s > 0; s >>= 1) {
        if (tid < s) red[tid] += red[tid + s];
        __syncthreads();
    }
    if (tid == 0) atomicAdd(&acc[1], red[0]);
}

__global__ void init_kernel(float* __restrict__ acc) {
    acc[0] = 0.f;
    acc[1] = 0.f;
}

__global__ void finalize_kernel(const float* __restrict__ acc,
                                float* __restrict__ out) {
    out[0] = acc[0] / (float)(B_ * N_) + acc[1] / (float)(B_ * M_);
}

// ---------------------------------------------------------------------------
extern "C" void kernel_launch(void* const* d_in, const int* in_sizes, int n_in,
                              void* d_out, int out_size, void* d_ws, size_t ws_size,
                              hipStream_t stream) {
    const float* pcl       = (const float*)d_in[0];
    const float* trans     = (const float*)d_in[1];
    const float* rot       = (const float*)d_in[2];
    const float* size      = (const float*)d_in[3];
    const float* shape_eps = (const float*)d_in[4];
    const float* deform    = (const float*)d_in[5];
    const float* probs     = (const float*)d_in[6];
    const float* etas      = (const float*)d_in[7];
    const float* omegas    = (const float*)d_in[8];
    float* out = (float*)d_out;

    float* ws    = (float*)d_ws;
    float* acc   = ws;                    // 2 floats
    float* dmin  = ws + 8;                // B*N*M floats (1 MB)
    float* meanS = dmin + (size_t)B_ * N_ * M_;  // B*M floats

    init_kernel<<<1, 1, 0, stream>>>(acc);
    sq_dist_kernel<<<B_ * M_, 256, 0, stream>>>(
        pcl, trans, rot, size, shape_eps, deform, etas, omegas, dmin, meanS);
    pcl2prim_kernel<<<(B_ * N_) / 256, 256, 0, stream>>>(dmin, probs, acc);
    prim2pcl_kernel<<<1, 128, 0, stream>>>(size, meanS, acc);
    finalize_kernel<<<1, 1, 0, stream>>>(acc, out);
}